// GCN_18133351924450
// MI455X (gfx1250) — compile-verified
//
#include <hip/hip_runtime.h>
#include <hip/hip_bf16.h>

typedef __attribute__((ext_vector_type(2))) float v2f;
typedef __attribute__((ext_vector_type(4))) float v4f;
typedef __attribute__((ext_vector_type(8))) float v8f;

#define FIN 256
#define HID 64
#define NC  16

// ---------------- CDNA5 async global->LDS helpers ----------------
// global_load_async_to_lds_b128: VDST = LDS byte address VGPR, VADDR = 64-bit global addr.
// Tracked with ASYNCcnt; wave waits with s_wait_asynccnt, then workgroup barrier.

__device__ __forceinline__ void async_copy_b128(unsigned lds_off, const void* gptr) {
  asm volatile("global_load_async_to_lds_b128 %0, %1, off"
               :: "v"(lds_off), "v"((unsigned long long)(size_t)gptr)
               : "memory");
}

__device__ __forceinline__ void wait_async0() {
  asm volatile("s_wait_asynccnt 0x0" ::: "memory");
}

// low 32 bits of a generic pointer to __shared__ = byte offset in the WG's LDS allocation
__device__ __forceinline__ unsigned lds_offset(const void* p) {
  return (unsigned)(size_t)p;
}

// ---------------- degree / normalization ----------------

__global__ void k_init_deg(float* deg, int N) {
  int i = blockIdx.x * blockDim.x + threadIdx.x;
  if (i < N) deg[i] = 1.0f;  // self-loop contributes 1 to every node's degree
}

__global__ void k_count_deg(const long long* __restrict__ row, float* deg, int E) {
  int i = blockIdx.x * blockDim.x + threadIdx.x;
  if (i < E) unsafeAtomicAdd(&deg[(int)row[i]], 1.0f);
}

__global__ void k_rsqrt_inplace(float* deg, int N) {
  int i = blockIdx.x * blockDim.x + threadIdx.x;
  if (i < N) deg[i] = rsqrtf(deg[i]);  // deg >= 1 always (self-loops)
}

// ---------------- GEMM1: h = x @ W1  (N x 256) @ (256 x 64), fp32 WMMA ----------------
// W1 (64 KB) staged in LDS via async copies; one wave per 16-row tile, 16x64 per wave.

__global__ void __launch_bounds__(128) k_gemm1(const float* __restrict__ x,
                                               const float* __restrict__ W1,
                                               float* __restrict__ h, int ntiles) {
  __shared__ float sW[FIN * HID];  // 64 KB

  // cooperative async stage of W1: 4096 x b128 chunks over 128 threads
  {
    const unsigned base = lds_offset(sW);
    const int tid = threadIdx.x;
#pragma unroll
    for (int i = 0; i < (FIN * HID / 4) / 128; ++i) {
      int c = tid + i * 128;
      async_copy_b128(base + (unsigned)c * 16u, W1 + (size_t)c * 4);
    }
    wait_async0();
  }
  __syncthreads();

  const int lane = threadIdx.x & 31;
  const int wave = threadIdx.x >> 5;
  const int mt = blockIdx.x * (blockDim.x >> 5) + wave;
  if (mt >= ntiles) return;  // wave-uniform: EXEC all-ones for WMMA below
  const int m0   = mt << 4;
  const int half = lane >> 4;   // selects K pair
  const int l16  = lane & 15;   // A: row m, B/C: col n

  const float* __restrict__ xrow = x + (size_t)(m0 + l16) * FIN;
  v8f acc0 = {}, acc1 = {}, acc2 = {}, acc3 = {};

#pragma unroll 4
  for (int k = 0; k < FIN; k += 4) {
    const int ka = k + 2 * half;
    const v2f a = *(const v2f*)(xrow + ka);          // A[m][ka], A[m][ka+1]
    const float* wp = sW + ka * HID + l16;           // W1[ka][n] from LDS
    v2f b0 = { wp[0],  wp[HID + 0]  };
    v2f b1 = { wp[16], wp[HID + 16] };
    v2f b2 = { wp[32], wp[HID + 32] };
    v2f b3 = { wp[48], wp[HID + 48] };
    acc0 = __builtin_amdgcn_wmma_f32_16x16x4_f32(false, a, false, b0, (short)0, acc0, false, false);
    acc1 = __builtin_amdgcn_wmma_f32_16x16x4_f32(false, a, false, b1, (short)0, acc1, false, false);
    acc2 = __builtin_amdgcn_wmma_f32_16x16x4_f32(false, a, false, b2, (short)0, acc2, false, false);
    acc3 = __builtin_amdgcn_wmma_f32_16x16x4_f32(false, a, false, b3, (short)0, acc3, false, false);
  }

#pragma unroll
  for (int v = 0; v < 8; ++v) {
    float* hp = h + (size_t)(m0 + v + 8 * half) * HID + l16;  // C: M = v + 8*half, N = l16
    hp[0]  = acc0[v];
    hp[16] = acc1[v];
    hp[32] = acc2[v];
    hp[48] = acc3[v];
  }
}

// ---------------- GEMM2: h2 = h1 @ W2  (N x 64) @ (64 x 16), fp32 WMMA ----------------

__global__ void __launch_bounds__(128) k_gemm2(const float* __restrict__ h1,
                                               const float* __restrict__ W2,
                                               float* __restrict__ h2, int ntiles) {
  __shared__ float sW[HID * NC];  // 4 KB

  {
    const unsigned base = lds_offset(sW);
    const int tid = threadIdx.x;
#pragma unroll
    for (int i = 0; i < (HID * NC / 4) / 128; ++i) {
      int c = tid + i * 128;
      async_copy_b128(base + (unsigned)c * 16u, W2 + (size_t)c * 4);
    }
    wait_async0();
  }
  __syncthreads();

  const int lane = threadIdx.x & 31;
  const int wave = threadIdx.x >> 5;
  const int mt = blockIdx.x * (blockDim.x >> 5) + wave;
  if (mt >= ntiles) return;
  const int m0   = mt << 4;
  const int half = lane >> 4;
  const int l16  = lane & 15;

  const float* __restrict__ arow = h1 + (size_t)(m0 + l16) * HID;
  v8f acc = {};

#pragma unroll
  for (int k = 0; k < HID; k += 4) {
    const int ka = k + 2 * half;
    const v2f a = *(const v2f*)(arow + ka);
    v2f b = { sW[ka * NC + l16], sW[(ka + 1) * NC + l16] };
    acc = __builtin_amdgcn_wmma_f32_16x16x4_f32(false, a, false, b, (short)0, acc, false, false);
  }

#pragma unroll
  for (int v = 0; v < 8; ++v)
    h2[(size_t)(m0 + v + 8 * half) * NC + l16] = acc[v];
}

// ---------------- self-loop init (vectorized):  agg[n,f] = h[n,f] * dinv[n]^2 ----------------

__global__ void k_self_init4(const float* __restrict__ h, const float* __restrict__ dinv,
                             float* __restrict__ agg, long long total4, int logQ) {
  long long t = (long long)blockIdx.x * blockDim.x + threadIdx.x;
  if (t >= total4) return;
  int n = (int)(t >> logQ);  // logQ = log2(F/4)
  float d = dinv[n];
  float s = d * d;
  v4f v = *(const v4f*)(h + t * 4);
  v4f o = { v.x * s, v.y * s, v.z * s, v.w * s };
  *(v4f*)(agg + t * 4) = o;
}

// ---------------- edge scatter (vectorized): agg[row] += h[col] * dinv[row]*dinv[col] ----------------
// One thread per (edge, 4-feature group): 1 b128 gather + 4 f32 atomics.

__global__ void k_scatter4(const long long* __restrict__ row, const long long* __restrict__ col,
                           const float* __restrict__ dinv, const float* __restrict__ h,
                           float* __restrict__ agg, long long total4, int logQ) {
  long long t = (long long)blockIdx.x * blockDim.x + threadIdx.x;
  if (t >= total4) return;
  const int Q = 1 << logQ;        // F/4
  int e = (int)(t >> logQ);
  int q = (int)(t & (Q - 1));
  int r = (int)row[e];
  int c = (int)col[e];
  float s = dinv[r] * dinv[c];
  v4f v = *(const v4f*)(h + ((size_t)c << (logQ + 2)) + q * 4);
  float* ap = agg + ((size_t)r << (logQ + 2)) + q * 4;
  unsafeAtomicAdd(ap + 0, v.x * s);
  unsafeAtomicAdd(ap + 1, v.y * s);
  unsafeAtomicAdd(ap + 2, v.z * s);
  unsafeAtomicAdd(ap + 3, v.w * s);
}

// ---------------- bias + ReLU in place (vectorized) ----------------

__global__ void k_bias_relu4(float* __restrict__ a, const float* __restrict__ b,
                             long long total4, int qmask) {
  long long t = (long long)blockIdx.x * blockDim.x + threadIdx.x;
  if (t >= total4) return;
  v4f bv = *(const v4f*)(b + (int)(t & qmask) * 4);
  v4f v = *(const v4f*)(a + t * 4);
  v4f o = { v.x + bv.x, v.y + bv.y, v.z + bv.z, v.w + bv.w };
  o.x = o.x > 0.0f ? o.x : 0.0f;
  o.y = o.y > 0.0f ? o.y : 0.0f;
  o.z = o.z > 0.0f ? o.z : 0.0f;
  o.w = o.w > 0.0f ? o.w : 0.0f;
  *(v4f*)(a + t * 4) = o;
}

// ---------------- bias + log_softmax in place (one thread per node, C=16) ----------------

__global__ void k_bias_logsoftmax(float* __restrict__ out, const float* __restrict__ b2, int N) {
  int n = blockIdx.x * blockDim.x + threadIdx.x;
  if (n >= N) return;
  float* p = out + (size_t)n * NC;
  float v[NC];
  float m = -1e30f;
#pragma unroll
  for (int i = 0; i < NC; ++i) { v[i] = p[i] + b2[i]; m = v[i] > m ? v[i] : m; }
  float s = 0.0f;
#pragma unroll
  for (int i = 0; i < NC; ++i) s += __expf(v[i] - m);
  float l = m + __logf(s);
#pragma unroll
  for (int i = 0; i < NC; ++i) p[i] = v[i] - l;
}

// ---------------- launcher ----------------

extern "C" void kernel_launch(void* const* d_in, const int* in_sizes, int n_in,
                              void* d_out, int out_size, void* d_ws, size_t ws_size,
                              hipStream_t stream) {
  const float*     x   = (const float*)d_in[0];
  const long long* ei  = (const long long*)d_in[1];   // int64 edge_index [2, E]
  const float*     W1  = (const float*)d_in[2];
  const float*     b1  = (const float*)d_in[3];
  const float*     W2  = (const float*)d_in[4];
  const float*     b2  = (const float*)d_in[5];
  float*           out = (float*)d_out;

  const int N = in_sizes[0] / FIN;     // 50000
  const int E = in_sizes[1] / 2;       // 1600000
  const long long* row = ei;           // targets
  const long long* col = ei + E;       // sources

  // workspace layout (floats): dinv[N] | h[N*64] | agg1[N*64] | h2[N*16]
  float* dinv = (float*)d_ws;
  float* h    = dinv + N;
  float* agg1 = h    + (size_t)N * HID;
  float* h2   = agg1 + (size_t)N * HID;

  const int ntiles = N / 16;           // 50000 % 16 == 0
  const int wpb = 4;                   // waves per block (blockDim 128)

  // 1) symmetric normalization dinv = deg^{-1/2}
  k_init_deg<<<(N + 255) / 256, 256, 0, stream>>>(dinv, N);
  k_count_deg<<<(E + 255) / 256, 256, 0, stream>>>(row, dinv, E);
  k_rsqrt_inplace<<<(N + 255) / 256, 256, 0, stream>>>(dinv, N);

  // 2) layer 1 dense transform (fp32 WMMA, W1 async-staged in LDS)
  k_gemm1<<<(ntiles + wpb - 1) / wpb, 32 * wpb, 0, stream>>>(x, W1, h, ntiles);

  // 3) layer 1 aggregation: self-loops then edges (F=64 -> Q=16, logQ=4)
  long long n64q = (long long)N * (HID / 4);
  long long e64q = (long long)E * (HID / 4);
  k_self_init4<<<(unsigned)((n64q + 255) / 256), 256, 0, stream>>>(h, dinv, agg1, n64q, 4);
  k_scatter4<<<(unsigned)((e64q + 255) / 256), 256, 0, stream>>>(row, col, dinv, h, agg1, e64q, 4);
  k_bias_relu4<<<(unsigned)((n64q + 255) / 256), 256, 0, stream>>>(agg1, b1, n64q, (HID / 4) - 1);

  // 4) layer 2 dense transform (fp32 WMMA, W2 async-staged in LDS)
  k_gemm2<<<(ntiles + wpb - 1) / wpb, 32 * wpb, 0, stream>>>(agg1, W2, h2, ntiles);

  // 5) layer 2 aggregation straight into d_out (F=16 -> Q=4, logQ=2)
  long long n16q = (long long)N * (NC / 4);
  long long e16q = (long long)E * (NC / 4);
  k_self_init4<<<(unsigned)((n16q + 255) / 256), 256, 0, stream>>>(h2, dinv, out, n16q, 2);
  k_scatter4<<<(unsigned)((e16q + 255) / 256), 256, 0, stream>>>(row, col, dinv, h2, out, e16q, 2);

  // 6) bias + log_softmax in place
  k_bias_logsoftmax<<<(N + 255) / 256, 256, 0, stream>>>(out, b2, N);
}